// PenaltyMethod_18176301597216
// MI455X (gfx1250) — compile-verified
//
#include <hip/hip_runtime.h>
#include <hip/hip_bf16.h>

#define B_ 4
#define V_ 4096
#define F_ 8192
#define P_ 6
#define E_ 2048

typedef __attribute__((ext_vector_type(2))) float v2f;
typedef __attribute__((ext_vector_type(8))) float v8f;

// ws layout: ws[0..3] = signed volume sums, ws[4..7] = chamfer raw sums
__global__ void pm_init_ws(float* ws) {
    int t = threadIdx.x;
    if (t < 8) ws[t] = 0.0f;
}

// One wave handles one (b, p, vtile of 16 points). Loops over 128 edge tiles,
// each a V_WMMA_F32_16X16X4_F32 computing a 16x16 tile of squared distances
// via the rank-4 expansion d2 = (u,w,|x|^2,1) . (-2y0,-2y1,1,|y|^2).
__global__ __launch_bounds__(256) void pm_chamfer_wmma(
        const float* __restrict__ xs,
        const float* __restrict__ proj,
        const float* __restrict__ edgemaps,
        float* __restrict__ ws_chamfer) {
    const int VT = V_ / 16;
    const int lane = threadIdx.x & 31;
    const int wave = threadIdx.x >> 5;
    const int gwave = blockIdx.x * (blockDim.x >> 5) + wave;
    const int vtile = gwave % VT;
    const int rem   = gwave / VT;
    const int p     = rem % P_;
    const int b     = rem / P_;

    const int half = lane >> 4;   // 0: K=0,1   1: K=2,3
    const int idx  = lane & 15;   // row (A) / col (B) within tile

    // ---- Build A tile: 16 projected points, K = (u, w, u^2+w^2, 1) ----
    const int v = vtile * 16 + idx;
    const float* xp = xs + ((size_t)b * V_ + v) * 3;
    const float x0 = xp[0], x1 = xp[1], x2 = xp[2];
    const float* M = proj + p * 12;
    const float p0 = M[0]*x0 + M[1]*x1 + M[2]*x2  + M[3];
    const float p1 = M[4]*x0 + M[5]*x1 + M[6]*x2  + M[7];
    const float p2 = M[8]*x0 + M[9]*x1 + M[10]*x2 + M[11];
    const float u  = p0 / p2;
    const float w  = p1 / p2;
    const float n2 = u * u + w * w;

    v2f a;
    a.x = half ? n2   : u;   // VGPR0: K0 (lanes 0-15) / K2 (lanes 16-31)
    a.y = half ? 1.0f : w;   // VGPR1: K1             / K3

    const float* em = edgemaps + ((size_t)(b * P_ + p)) * E_ * 2;

    v8f mind;
#pragma unroll
    for (int i = 0; i < 8; ++i) mind[i] = 3.4e38f;
    v8f zero = {};

    for (int et = 0; et < E_ / 16; ++et) {
        const int e = et * 16 + idx;
        const float y0 = em[2 * e + 0];
        const float y1 = em[2 * e + 1];
        v2f bm;
        bm.x = half ? 1.0f              : -2.0f * y0;  // K0 / K2
        bm.y = half ? (y0*y0 + y1*y1)   : -2.0f * y1;  // K1 / K3
        // D = A x B + 0  (16x16 f32 tile of squared distances)
        v8f d = __builtin_amdgcn_wmma_f32_16x16x4_f32(
            /*neg_a=*/false, a, /*neg_b=*/false, bm,
            /*c_mod=*/(short)0, zero, /*reuse_a=*/false, /*reuse_b=*/false);
#pragma unroll
        for (int i = 0; i < 8; ++i) mind[i] = fminf(mind[i], d[i]);
    }

    // Min over columns: XOR-shuffle reduction within each 16-lane half
    // (cols of a row live in one VGPR across lanes 0-15 or 16-31).
#pragma unroll
    for (int off = 1; off < 16; off <<= 1) {
#pragma unroll
        for (int i = 0; i < 8; ++i) {
            float o = __shfl_xor(mind[i], off, 32);
            mind[i] = fminf(mind[i], o);
        }
    }

    // lane 0 holds row-mins for M=0..7, lane 16 for M=8..15
    if (idx == 0) {
        float s = 0.0f;
#pragma unroll
        for (int i = 0; i < 8; ++i) s += mind[i];
        atomicAdd(&ws_chamfer[b], s);
    }
}

__global__ __launch_bounds__(256) void pm_volume(
        const float* __restrict__ xs,
        const int* __restrict__ faces,
        float* __restrict__ ws_vol) {
    const int b = blockIdx.y;
    const int f = blockIdx.x * blockDim.x + threadIdx.x;
    float fv = 0.0f;
    if (f < F_) {
        const int* fc = faces + ((size_t)b * F_ + f) * 3;
        const int i0 = fc[0], i1 = fc[1], i2 = fc[2];
        const float* a0 = xs + ((size_t)b * V_ + i0) * 3;
        const float* a1 = xs + ((size_t)b * V_ + i1) * 3;
        const float* a2 = xs + ((size_t)b * V_ + i2) * 3;
        const float c0 = a0[1]*a1[2] - a0[2]*a1[1];
        const float c1 = a0[2]*a1[0] - a0[0]*a1[2];
        const float c2 = a0[0]*a1[1] - a0[1]*a1[0];
        fv = (c0*a2[0] + c1*a2[1] + c2*a2[2]) * (1.0f / 6.0f);
    }
    __shared__ float red[256];
    red[threadIdx.x] = fv;
    __syncthreads();
    for (int s = 128; s > 0; s >>= 1) {
        if (threadIdx.x < s) red[threadIdx.x] += red[threadIdx.x + s];
        __syncthreads();
    }
    if (threadIdx.x == 0) atomicAdd(&ws_vol[b], red[0]);
}

__global__ void pm_finalize(const float* __restrict__ ws,
                            const float* __restrict__ target,
                            float* __restrict__ out) {
    const int b = threadIdx.x;
    if (b < B_) {
        out[b] = ws[4 + b] * (1.0f / ((float)P_ * (float)V_));
        const float diff = fabsf(ws[b]) - target[b];
        out[4 + b] = diff * diff;
    }
}

extern "C" void kernel_launch(void* const* d_in, const int* in_sizes, int n_in,
                              void* d_out, int out_size, void* d_ws, size_t ws_size,
                              hipStream_t stream) {
    const float* xs       = (const float*)d_in[0];
    const float* proj     = (const float*)d_in[1];
    const float* edgemaps = (const float*)d_in[2];
    const int*   faces    = (const int*)d_in[3];
    const float* target   = (const float*)d_in[4];
    float* out = (float*)d_out;
    float* ws  = (float*)d_ws;

    pm_init_ws<<<1, 32, 0, stream>>>(ws);

    const int nwaves = B_ * P_ * (V_ / 16);   // 6144 waves
    const int wpb = 8;                        // 256 threads/block = 8 wave32
    pm_chamfer_wmma<<<nwaves / wpb, 256, 0, stream>>>(xs, proj, edgemaps, ws + 4);

    dim3 vg(F_ / 256, B_);
    pm_volume<<<vg, 256, 0, stream>>>(xs, faces, ws);

    pm_finalize<<<1, 32, 0, stream>>>(ws, target, out);
}